// ConflictDetector_78881369358557
// MI455X (gfx1250) — compile-verified
//
#include <hip/hip_runtime.h>
#include <hip/hip_bf16.h>

typedef __attribute__((ext_vector_type(2))) float v2f;
typedef __attribute__((ext_vector_type(8))) float v8f;

#define NPTS 2048
#define ED   64

// ---------------------------------------------------------------------------
// Kernel 1: per-row embedding gather + projection + split MLP-in.
//   e   = concat(subj,rel,obj) @ proj_w^T + proj_b          (N x 64)
//   hi' = e @ w1[:, :64]^T + b1   (b1 folded here)          (N x 64)
//   hj  = e @ w1[:, 64:]^T                                  (N x 64)
// One 64-thread block per row; ~40M MACs total -> negligible.
// ---------------------------------------------------------------------------
__global__ void precompute_kernel(const int* __restrict__ subj_idx,
                                  const int* __restrict__ rel_idx,
                                  const int* __restrict__ obj_idx,
                                  const float* __restrict__ subj_table,
                                  const float* __restrict__ rel_table,
                                  const float* __restrict__ obj_table,
                                  const float* __restrict__ proj_w,
                                  const float* __restrict__ proj_b,
                                  const float* __restrict__ w1,
                                  const float* __restrict__ b1,
                                  float* __restrict__ hiOut,
                                  float* __restrict__ hjOut)
{
    __shared__ float comb[192];
    __shared__ float e[64];
    const int n = blockIdx.x;
    const int t = threadIdx.x;            // 0..63

    const int si = subj_idx[n], ri = rel_idx[n], oi = obj_idx[n];
    comb[t]       = subj_table[si * ED + t];
    comb[64 + t]  = rel_table[ri * ED + t];
    comb[128 + t] = obj_table[oi * ED + t];
    __syncthreads();

    float acc = proj_b[t];
    const float* pw = proj_w + t * 192;   // proj_w is (64,192) row-major
    #pragma unroll 8
    for (int k = 0; k < 192; ++k) acc += pw[k] * comb[k];
    e[t] = acc;
    __syncthreads();

    float hi = b1[t];                     // fold b1 into hi
    float hj = 0.0f;
    const float* w1r = w1 + t * 128;      // w1 is (64,128) row-major
    #pragma unroll 8
    for (int k = 0; k < 64; ++k) {
        hi += e[k] * w1r[k];
        hj += e[k] * w1r[64 + k];
    }
    hiOut[n * ED + t] = hi;
    hjOut[n * ED + t] = hj;
}

// ---------------------------------------------------------------------------
// Kernel 2: pairwise scoring via V_WMMA_F32_16X16X4_F32.
// GEMM view per wave: M = 16 pairs (one i, a 16-wide j-group), K = 64, N = 32.
// A[M][k] = relu(hi'_i[k] + hj_{j0+M}[k]) built per K-step in registers.
// B[k][n] = w2[n][k] preloaded into 64 VGPRs (reused across 32 i-iterations).
// A 16x4 fp32 layout (ISA 7.12.2): lane&15 = M, lane>>4 selects K pair.
// triu(k=1) exploited: fully-masked blocks / i-rows skip all WMMA work
// via wave-uniform branches (EXEC stays all-ones on the WMMA path).
// ---------------------------------------------------------------------------
__launch_bounds__(128, 1)
__global__ void pair_score_kernel(const float* __restrict__ hiG,
                                  const float* __restrict__ hjG,
                                  const float* __restrict__ w2,   // (32,64)
                                  const float* __restrict__ b2,   // (32)
                                  const float* __restrict__ w3,   // (1,32)
                                  const float* __restrict__ b3,   // (1)
                                  float* __restrict__ out)        // (2048,2048)
{
    __shared__ float ldsHi[128 * ED];     // 32 KB (of 320 KB WGP LDS)

    const int lane = threadIdx.x & 31;
    const int wave = threadIdx.x >> 5;    // 4 waves / block
    const int nloc = lane & 15;           // N (channel) / M (pair) sub-index
    const int par  = lane >> 4;           // K-pair / M-half selector
    const int j0     = blockIdx.x * 16;   // 16-wide j-group
    const int iBlock = blockIdx.y * 128;  // 128 i's per block

    // --- Fast path: whole block below/on the diagonal -> zero-fill, done. ---
    // (j in [j0, j0+15], i in [iBlock, iBlock+127]; all masked iff j0+15 <= iBlock)
    if (j0 + 15 <= iBlock) {
        const int iStartZ = iBlock + wave * 32;
        if (par == 0) {                   // 16 lanes cover the 16 j's of a row
            for (int il = 0; il < 32; ++il) {
                out[(size_t)(iStartZ + il) * NPTS + j0 + nloc] = 0.0f;
            }
        }
        return;                           // uniform: whole block exits
    }

    // Cooperative load of the hi' tile for this block's 128 i's.
    #pragma unroll 4
    for (int r = 0; r < 64; ++r) {
        const int idx = threadIdx.x + r * 128;
        ldsHi[idx] = hiG[iBlock * ED + idx];
    }
    __syncthreads();

    // Per-lane hj fragment values: lane handles j = j0 + nloc, K subset per par.
    float hjr[32];
    {
        const float* hjrow = hjG + (j0 + nloc) * ED + 2 * par;
        #pragma unroll
        for (int s = 0; s < 16; ++s) {
            hjr[2 * s]     = hjrow[4 * s];
            hjr[2 * s + 1] = hjrow[4 * s + 1];
        }
    }

    // Preload B fragments: B[k][n] = w2[n][k]; two N-halves.
    v2f blo[16], bhi[16];
    #pragma unroll
    for (int s = 0; s < 16; ++s) {
        const int k = 4 * s + 2 * par;
        blo[s].x = w2[nloc * ED + k];
        blo[s].y = w2[nloc * ED + k + 1];
        bhi[s].x = w2[(nloc + 16) * ED + k];
        bhi[s].y = w2[(nloc + 16) * ED + k + 1];
    }
    const float b2lo = b2[nloc],  b2hi = b2[nloc + 16];
    const float w3lo = w3[nloc],  w3hi = w3[nloc + 16];
    const float b3s  = b3[0];

    const int iStart = wave * 32;
    for (int il = 0; il < 32; ++il) {
        const int iloc = iStart + il;
        const int i    = iBlock + iloc;

        // --- Fast path: this whole i-row of the j-group is masked. ---
        // Wave-uniform branch (i, j0 uniform) -> EXEC untouched for WMMA path.
        if (j0 + 15 <= i) {
            if (par == 0) out[(size_t)i * NPTS + j0 + nloc] = 0.0f;
            continue;
        }

        const float* hrow = &ldsHi[iloc * ED + 2 * par];  // broadcast reads

        v8f acc0 = {};   // channels 0..15
        v8f acc1 = {};   // channels 16..31
        #pragma unroll
        for (int s = 0; s < 16; ++s) {
            v2f a;
            const float h0 = hrow[4 * s]     + hjr[2 * s];
            const float h1 = hrow[4 * s + 1] + hjr[2 * s + 1];
            a.x = h0 > 0.0f ? h0 : 0.0f;
            a.y = h1 > 0.0f ? h1 : 0.0f;
            acc0 = __builtin_amdgcn_wmma_f32_16x16x4_f32(
                       false, a, false, blo[s], (short)0, acc0, false, false);
            acc1 = __builtin_amdgcn_wmma_f32_16x16x4_f32(
                       false, a, false, bhi[s], (short)0, acc1, false, false);
        }

        // Epilogue: relu(+b2), dot with w3 across 32 channels (lane dim),
        // sigmoid, triu(k=1) mask, store. D layout: lane&15 = N,
        // VGPR m holds pair M=m (lanes 0-15) / M=m+8 (lanes 16-31).
        #pragma unroll
        for (int m = 0; m < 8; ++m) {
            float t0 = acc0[m] + b2lo; t0 = t0 > 0.0f ? t0 : 0.0f;
            float t1 = acc1[m] + b2hi; t1 = t1 > 0.0f ? t1 : 0.0f;
            float t = t0 * w3lo + t1 * w3hi;
            t += __shfl_xor(t, 1, 32);
            t += __shfl_xor(t, 2, 32);
            t += __shfl_xor(t, 4, 32);
            t += __shfl_xor(t, 8, 32);    // xor<16 stays within the 16-lane half
            const int j = j0 + m + 8 * par;
            float score = 1.0f / (1.0f + __expf(-(t + b3s)));
            if (j <= i) score = 0.0f;     // triu(k=1)
            if (nloc == 0) out[(size_t)i * NPTS + j] = score;
        }
    }
}

extern "C" void kernel_launch(void* const* d_in, const int* in_sizes, int n_in,
                              void* d_out, int out_size, void* d_ws, size_t ws_size,
                              hipStream_t stream) {
    const int*   subj_idx   = (const int*)  d_in[0];
    const int*   rel_idx    = (const int*)  d_in[1];
    const int*   obj_idx    = (const int*)  d_in[2];
    const float* subj_table = (const float*)d_in[3];
    const float* rel_table  = (const float*)d_in[4];
    const float* obj_table  = (const float*)d_in[5];
    const float* proj_w     = (const float*)d_in[6];
    const float* proj_b     = (const float*)d_in[7];
    const float* w1         = (const float*)d_in[8];
    const float* b1         = (const float*)d_in[9];
    const float* w2         = (const float*)d_in[10];
    const float* b2         = (const float*)d_in[11];
    const float* w3         = (const float*)d_in[12];
    const float* b3         = (const float*)d_in[13];
    float* out = (float*)d_out;

    float* hiB = (float*)d_ws;            // 2048*64 f32 = 512 KB
    float* hjB = hiB + NPTS * ED;         // 2048*64 f32 = 512 KB

    precompute_kernel<<<NPTS, 64, 0, stream>>>(
        subj_idx, rel_idx, obj_idx, subj_table, rel_table, obj_table,
        proj_w, proj_b, w1, b1, hiB, hjB);

    dim3 grid(NPTS / 16, NPTS / 128);     // (128 j-groups, 16 i-blocks)
    pair_score_kernel<<<grid, 128, 0, stream>>>(hiB, hjB, w2, b2, w3, b3, out);
}